// AnomalyMapGenerator_10892037063016
// MI455X (gfx1250) — compile-verified
//
#include <hip/hip_runtime.h>

typedef __attribute__((ext_vector_type(2))) float v2f;
typedef __attribute__((ext_vector_type(8))) float v8f;

#define KS   33
#define HPAD 16
#define PH   28      // patch side
#define IMG  512     // output side
#define KP   32      // padded K dim
#define NB   32      // batch

// workspace layout (float offsets)
#define WPAD_OFF 0                        // Wpad  [512][32]
#define SPAD_OFF (IMG * KP)               // S^T   [32][32][32]  (ST[b][n][k] = S[b][k][n])
#define T_OFF    (SPAD_OFF + NB * KP * KP)// T     [32][512][32]

// ---------------------------------------------------------------------------
// Kernel 0: build fused upsample+blur operator Wpad and padded/transposed S.
// W[y][p] = sum_t k1[t] * [0 <= s=y+t-16 < 512  and  (s*28)>>9 == p]
// k1[t] = blur_row16[t] / sum(blur_row16)   (since k2d = k1 outer k1)
// ---------------------------------------------------------------------------
__global__ void AMG_prep(const float* __restrict__ patch,
                         const float* __restrict__ blur,
                         float* ws) {
  int gid = blockIdx.x * blockDim.x + threadIdx.x;
  const float* krow = blur + HPAD * KS;  // center row of 33x33 kernel

  if (gid < IMG * KP) {
    int y = gid / KP, p = gid % KP;
    float val = 0.0f;
    if (p < PH) {
      float sum = 0.0f;
      #pragma unroll
      for (int j = 0; j < KS; ++j) sum += krow[j];
      float acc = 0.0f;
      #pragma unroll
      for (int t = 0; t < KS; ++t) {
        int s = y + t - HPAD;
        if (s >= 0 && s < IMG && ((s * PH) >> 9) == p) acc += krow[t];
      }
      val = acc / sum;
    }
    ws[WPAD_OFF + gid] = val;
  } else {
    int sid = gid - IMG * KP;            // covers NB*KP*KP exactly
    int b = sid / (KP * KP);
    int r = sid % (KP * KP);
    int n = r / KP;                      // column of S
    int k = r % KP;                      // row of S
    float v = 0.0f;
    if (k < PH && n < PH) v = patch[b * PH * PH + k * PH + n];
    ws[SPAD_OFF + sid] = v;              // transposed + zero-padded
  }
}

// ---------------------------------------------------------------------------
// WMMA helper: one 16x16 fp32 tile, K=32 in 8 steps of V_WMMA_F32_16X16X4_F32.
// A operand: lane (0..15 -> M, 16..31 -> M, K+2), float2 = K,K+1  -> rowA[k..k+1]
// B operand: same striping with N in lane -> needs B^T rows contiguous in K.
// ---------------------------------------------------------------------------
__device__ __forceinline__ v8f wmma_tile_k32(const float* __restrict__ aRow,
                                             const float* __restrict__ bRow) {
  v8f acc = {};
  #pragma unroll
  for (int k0 = 0; k0 < KP; k0 += 4) {
    v2f a = *(const v2f*)(aRow + k0);
    v2f b = *(const v2f*)(bRow + k0);
    acc = __builtin_amdgcn_wmma_f32_16x16x4_f32(
        /*neg_a=*/false, a, /*neg_b=*/false, b,
        /*c_mod=*/(short)0, acc, /*reuse_a=*/false, /*reuse_b=*/false);
  }
  return acc;
}

// ---------------------------------------------------------------------------
// Stage A: T_b[512][32] = Wpad[512][32(K=28)] @ S_b[28][28]  (padded)
// 2048 waves: b(32) x mt(32) x nt(2), one 16x16 tile per wave.
// ---------------------------------------------------------------------------
__global__ void AMG_stageA(float* ws) {
  int wave = (blockIdx.x * blockDim.x + threadIdx.x) >> 5;
  int lane = threadIdx.x & 31;
  int nt = wave & 1;
  int mt = (wave >> 1) & 31;
  int b  = wave >> 6;

  int half = lane >> 4;
  int idx  = lane & 15;
  int kofs = 2 * half;

  const float* W  = ws + WPAD_OFF;
  const float* ST = ws + SPAD_OFF + b * KP * KP;  // ST[n][k] = S[k][n]
  float*       T  = ws + T_OFF    + b * IMG * KP;

  int arow = mt * 16 + idx;   // M row in W
  int bcol = nt * 16 + idx;   // N col in S  -> row of ST

  v8f acc = wmma_tile_k32(W + arow * KP + kofs, ST + bcol * KP + kofs);

  #pragma unroll
  for (int r = 0; r < 8; ++r) {
    int row = mt * 16 + r + 8 * half;     // C/D layout: lanes 16..31 -> M+8
    T[row * KP + bcol] = acc[r];
  }
}

// ---------------------------------------------------------------------------
// Stage B: out_b[512][512] = T_b[512][32] @ Wpad^T[32][512]
// B[k][n] = W[n][k] -> B operand rows are W rows (contiguous in K).
// 32768 waves: b(32) x mt(32) x nt(32).
// ---------------------------------------------------------------------------
__global__ void AMG_stageB(const float* __restrict__ ws,
                           float* __restrict__ out) {
  int wave = (blockIdx.x * blockDim.x + threadIdx.x) >> 5;
  int lane = threadIdx.x & 31;
  int nt = wave & 31;
  int mt = (wave >> 5) & 31;
  int b  = wave >> 10;

  int half = lane >> 4;
  int idx  = lane & 15;
  int kofs = 2 * half;

  const float* T = ws + T_OFF + b * IMG * KP;
  const float* W = ws + WPAD_OFF;

  int arow = mt * 16 + idx;
  int brow = nt * 16 + idx;

  v8f acc = wmma_tile_k32(T + arow * KP + kofs, W + brow * KP + kofs);

  float* ob = out + (size_t)b * IMG * IMG;
  #pragma unroll
  for (int r = 0; r < 8; ++r) {
    int row = mt * 16 + r + 8 * half;
    ob[row * IMG + nt * 16 + idx] = acc[r];
  }
}

// ---------------------------------------------------------------------------
extern "C" void kernel_launch(void* const* d_in, const int* in_sizes, int n_in,
                              void* d_out, int out_size, void* d_ws, size_t ws_size,
                              hipStream_t stream) {
  const float* patch = (const float*)d_in[0];  // [32,1,28,28]
  const float* blur  = (const float*)d_in[1];  // [1,1,33,33]
  float* out = (float*)d_out;                  // [32,512,512]
  float* ws  = (float*)d_ws;                   // needs ~2.25 MB

  // prep: 512*32 + 32*32*32 = 49152 items
  AMG_prep<<<192, 256, 0, stream>>>(patch, blur, ws);
  // stage A: 2048 waves = 256 blocks * 8 waves
  AMG_stageA<<<256, 256, 0, stream>>>(ws);
  // stage B: 32768 waves = 4096 blocks * 8 waves
  AMG_stageB<<<4096, 256, 0, stream>>>(ws, out);
}